// ClusteringLayer_82575041233210
// MI455X (gfx1250) — compile-verified
//
#include <hip/hip_runtime.h>
#include <hip/hip_bf16.h>

typedef float v2f __attribute__((ext_vector_type(2)));
typedef float v8f __attribute__((ext_vector_type(8)));

#define B_ 16
#define S_ 8192
#define D_ 128
#define C_ 64
#define NW 8                 // waves per block (256 threads, wave32)
#define TPB (NW * 32)
#define S_CHUNK 512          // rows of S per block -> 16 chunks per batch
#define CHUNKS_PER_B (S_ / S_CHUNK)
#define CN_STRIDE 132        // padded LDS stride (floats) -> conflict-free
#define TILE_STRIDE 132

// ---------------- Kernel 1: L2-normalize centroids into workspace ----------
__global__ __launch_bounds__(D_) void norm_centroids_k(
    const float* __restrict__ cent, float* __restrict__ cn_g) {
  __shared__ float red[D_];
  const int c = blockIdx.x, t = threadIdx.x;
  const float x = cent[c * D_ + t];
  red[t] = x * x;
  __syncthreads();
  for (int off = D_ / 2; off > 0; off >>= 1) {
    if (t < off) red[t] += red[t + off];
    __syncthreads();
  }
  const float inv = 1.0f / fmaxf(sqrtf(red[0]), 1e-12f);
  cn_g[c * D_ + t] = x * inv;
}

// ---------------- Kernel 2: WMMA sims + argmax -> assignments --------------
// Each wave: stage 16x128 raw keys to LDS, per-row inv-norm via shuffles,
// 16x64 sims via v_wmma_f32_16x16x4_f32 (K=128 in 32 steps, 4 N-tiles),
// butterfly argmax over 64 clusters per row.
__global__ __launch_bounds__(TPB) void assign_k(
    const float* __restrict__ keys, const float* __restrict__ mask,
    const float* __restrict__ cn_g, int* __restrict__ asn_g,
    float* __restrict__ out_asn) {
  extern __shared__ float smem[];
  float* l_cn = smem;                              // C_ * CN_STRIDE
  float* l_tiles = smem + C_ * CN_STRIDE;          // NW * 16 * TILE_STRIDE

  const int tid = threadIdx.x;
  const int wv = tid >> 5;
  const int lane = tid & 31;
  const int m = lane & 15;
  const int half = lane >> 4;

  const int b = blockIdx.x >> 4;                   // CHUNKS_PER_B == 16
  const int chunk = blockIdx.x & 15;
  const long base = (long)b * S_ + (long)chunk * S_CHUNK;

  // load normalized centroids into padded LDS (conflict-free stride)
  for (int i = tid; i < C_ * D_; i += TPB) {
    l_cn[(i >> 7) * CN_STRIDE + (i & 127)] = cn_g[i];
  }
  __syncthreads();

  float* tile = l_tiles + wv * (16 * TILE_STRIDE);

  for (int t = wv; t < S_CHUNK / 16; t += NW) {
    const long srow0 = base + (long)t * 16;

    // ---- stage 16x128 raw keys, compute per-row 1/||row|| -----------------
    float inv_m = 0.0f;  // inv-norm of row (lane&15), valid in both halves
    for (int i = 0; i < 16; ++i) {
      const float4 k4 =
          *(const float4*)(keys + (srow0 + i) * D_ + lane * 4);
      *(float4*)(tile + i * TILE_STRIDE + lane * 4) = k4;
      float ss = k4.x * k4.x + k4.y * k4.y + k4.z * k4.z + k4.w * k4.w;
      ss += __shfl_xor(ss, 16, 32);
      ss += __shfl_xor(ss, 8, 32);
      ss += __shfl_xor(ss, 4, 32);
      ss += __shfl_xor(ss, 2, 32);
      ss += __shfl_xor(ss, 1, 32);
      if (m == i) inv_m = 1.0f / fmaxf(sqrtf(ss), 1e-12f);
    }

    // ---- sims: 16x64 = 4 tiles of v_wmma_f32_16x16x4_f32, K=128 ----------
    v8f acc0 = {}, acc1 = {}, acc2 = {}, acc3 = {};
    const float* arow = tile + m * TILE_STRIDE;
#pragma unroll
    for (int kk = 0; kk < 32; ++kk) {
      const int k0 = kk * 4 + half * 2;
      v2f a;
      a.x = arow[k0] * inv_m;
      a.y = arow[k0 + 1] * inv_m;
      v2f b0, b1, b2, b3;
      const float* cp0 = l_cn + (0 + m) * CN_STRIDE + k0;
      const float* cp1 = l_cn + (16 + m) * CN_STRIDE + k0;
      const float* cp2 = l_cn + (32 + m) * CN_STRIDE + k0;
      const float* cp3 = l_cn + (48 + m) * CN_STRIDE + k0;
      b0.x = cp0[0]; b0.y = cp0[1];
      b1.x = cp1[0]; b1.y = cp1[1];
      b2.x = cp2[0]; b2.y = cp2[1];
      b3.x = cp3[0]; b3.y = cp3[1];
      acc0 = __builtin_amdgcn_wmma_f32_16x16x4_f32(false, a, false, b0,
                                                   (short)0, acc0, false, false);
      acc1 = __builtin_amdgcn_wmma_f32_16x16x4_f32(false, a, false, b1,
                                                   (short)0, acc1, false, false);
      acc2 = __builtin_amdgcn_wmma_f32_16x16x4_f32(false, a, false, b2,
                                                   (short)0, acc2, false, false);
      acc3 = __builtin_amdgcn_wmma_f32_16x16x4_f32(false, a, false, b3,
                                                   (short)0, acc3, false, false);
    }

    // ---- argmax over 64 clusters per row (first-max tie-break) -----------
    // C/D layout: VGPR r -> row r (lanes 0-15), row r+8 (lanes 16-31); N = lane&15.
#pragma unroll
    for (int r = 0; r < 8; ++r) {
      float bv = acc0[r];
      int bn = m;
      float v;
      v = acc1[r]; if (v > bv) { bv = v; bn = 16 + m; }
      v = acc2[r]; if (v > bv) { bv = v; bn = 32 + m; }
      v = acc3[r]; if (v > bv) { bv = v; bn = 48 + m; }
#pragma unroll
      for (int off = 1; off < 16; off <<= 1) {
        const float ov = __shfl_xor(bv, off, 32);
        const int on = __shfl_xor(bn, off, 32);
        if (ov > bv || (ov == bv && on < bn)) { bv = ov; bn = on; }
      }
      if (m == 0) {  // lanes 0 and 16 hold results for rows r and r+8
        const long srow = srow0 + r + half * 8;
        const float mv = mask[srow];
        asn_g[srow] = (mv != 0.0f) ? bn : -1;            // sentinel: excluded
        out_asn[srow] = (float)((mv != 0.0f) ? bn : 0);  // reference emits 0
      }
    }
  }
}

// ---------------- Kernel 3: deterministic segment-mean ---------------------
// One block per (b, c); assignments for batch b cached in LDS; each thread
// owns one of the 128 feature elements; rows accumulated in ascending s.
__global__ __launch_bounds__(D_) void reduce_k(
    const float* __restrict__ keys, const float* __restrict__ values,
    const float* __restrict__ cent, const int* __restrict__ asn_g,
    float* __restrict__ out_cc, float* __restrict__ out_cv) {
  __shared__ int lasn[S_];
  const int b = blockIdx.x >> 6;  // C_ == 64
  const int c = blockIdx.x & 63;
  const int e = threadIdx.x;

  for (int i = e; i < S_; i += D_) lasn[i] = asn_g[(long)b * S_ + i];
  __syncthreads();

  const float* kb = keys + (long)b * S_ * D_;
  const float* vb = values + (long)b * S_ * D_;
  float ks = 0.0f, vs = 0.0f, cnt = 0.0f;
  for (int s = 0; s < S_; ++s) {
    if (lasn[s] == c) {
      ks += kb[(long)s * D_ + e];
      vs += vb[(long)s * D_ + e];
      cnt += 1.0f;
    }
  }
  const long o = ((long)b * C_ + c) * D_ + e;
  if (cnt > 0.0f) {
    out_cc[o] = ks / cnt;
    out_cv[o] = vs / cnt;
  } else {
    out_cc[o] = cent[c * D_ + e];  // fallback to ORIGINAL centroids
    out_cv[o] = 0.0f;
  }
}

// ---------------- Host launcher -------------------------------------------
extern "C" void kernel_launch(void* const* d_in, const int* in_sizes, int n_in,
                              void* d_out, int out_size, void* d_ws,
                              size_t ws_size, hipStream_t stream) {
  const float* keys = (const float*)d_in[0];    // [B,S,D]
  const float* values = (const float*)d_in[1];  // [B,S,D]
  const float* mask = (const float*)d_in[2];    // [B,S]
  const float* cent = (const float*)d_in[3];    // [C,D]

  float* out = (float*)d_out;
  float* out_cc = out;                       // [B,C,D]
  float* out_cv = out + B_ * C_ * D_;        // [B,C,D]
  float* out_asn = out + 2 * B_ * C_ * D_;   // [B,S] (as float)

  float* cn_g = (float*)d_ws;                                  // C*D floats
  int* asn_g = (int*)((char*)d_ws + C_ * D_ * sizeof(float));  // B*S ints

  norm_centroids_k<<<C_, D_, 0, stream>>>(cent, cn_g);

  const size_t shmem =
      (size_t)(C_ * CN_STRIDE + NW * 16 * TILE_STRIDE) * sizeof(float);
  assign_k<<<B_ * CHUNKS_PER_B, TPB, shmem, stream>>>(keys, mask, cn_g, asn_g,
                                                      out_asn);

  reduce_k<<<B_ * C_, D_, 0, stream>>>(keys, values, cent, asn_g, out_cc,
                                       out_cv);
}